// CenterLoss_49400713839356
// MI455X (gfx1250) — compile-verified
//
#include <hip/hip_runtime.h>
#include <hip/hip_bf16.h>

#define B_ 128
#define T_ 80
#define C_ 6625
#define D_ 96
#define N_ (B_ * T_)   // 10240 rows

typedef __attribute__((ext_vector_type(2))) float v2f;
typedef __attribute__((ext_vector_type(8))) float v8f;

// ---------------------------------------------------------------------------
// Kernel 1: per-row argmax over C=6625 logits. One wave32 per row, coalesced
// b32 streaming with 4 loads in flight. First-occurrence tie-break to match
// jnp.argmax semantics. This kernel is the bandwidth-critical path (~271 MB).
// ---------------------------------------------------------------------------
__global__ __launch_bounds__(256) void cl_argmax_kernel(
    const float* __restrict__ logits, int* __restrict__ labels) {
    const int lane = threadIdx.x & 31;
    const int wave = threadIdx.x >> 5;
    const int n = blockIdx.x * 8 + wave;           // grid = N/8 blocks, all full

    const float* __restrict__ row = logits + (size_t)n * C_;
    float bv = -__builtin_inff();
    int   bi = 0;

    int i = lane;
    // main loop: 4 independent coalesced loads per iteration (128 elems/wave)
    for (; i + 96 < C_; i += 128) {
        float v0 = row[i];
        float v1 = row[i + 32];
        float v2 = row[i + 64];
        float v3 = row[i + 96];
        if (v0 > bv) { bv = v0; bi = i; }
        if (v1 > bv) { bv = v1; bi = i + 32; }
        if (v2 > bv) { bv = v2; bi = i + 64; }
        if (v3 > bv) { bv = v3; bi = i + 96; }
    }
    for (; i < C_; i += 32) {
        float v = row[i];
        if (v > bv) { bv = v; bi = i; }
    }
    // cross-lane (value,index) max-reduce, smaller index wins ties
#pragma unroll
    for (int off = 16; off > 0; off >>= 1) {
        float ov = __shfl_xor(bv, off, 32);
        int   oi = __shfl_xor(bi, off, 32);
        if (ov > bv || (ov == bv && oi < bi)) { bv = ov; bi = oi; }
    }
    if (lane == 0) labels[n] = bi;
}

// ---------------------------------------------------------------------------
// Kernel 2: per-row clipped squared distance to the argmax center.
// One wave32 per 16 rows. The 16 gathered dot products f_n . c_{label_n}
// are computed as the DIAGONAL of a 16x16 WMMA product:
//   A (16x4)  = feats rows n0..n0+15, K-slice           (2 VGPRs/step)
//   B (4x16)  = gathered center rows (col j = label of row n0+j), K-slice
//   acc (16x16 f32, 8 VGPRs) accumulated over 24 chained V_WMMA_F32_16X16X4_F32
// ||f||^2 and ||c||^2 are accumulated in f64 from the same loaded values
// (lane m holds K%4 in {0,1}, lane m+16 holds K%4 in {2,3}; xor-16 combines).
// EXEC is all-ones throughout the WMMA chain (divergence only afterwards).
// ---------------------------------------------------------------------------
__global__ __launch_bounds__(256) void cl_dist_wmma_kernel(
    const float* __restrict__ feats, const float* __restrict__ centers,
    const int* __restrict__ labels, double* __restrict__ partial) {
    const int lane  = threadIdx.x & 31;
    const int wave  = threadIdx.x >> 5;
    const int g     = blockIdx.x * 8 + wave;       // group of 16 rows
    const int n0    = g * 16;
    const int m     = lane & 15;                   // row within group
    const int khalf = lane >> 4;                   // 0: K%4 in {0,1}, 1: {2,3}

    const int lab = labels[n0 + m];
    const float* __restrict__ pf = feats   + (size_t)(n0 + m) * D_ + khalf * 2;
    const float* __restrict__ pc = centers + (size_t)lab       * D_ + khalf * 2;

    v8f acc = {};
    double sqf = 0.0, sqc = 0.0;
#pragma unroll
    for (int k0 = 0; k0 < D_; k0 += 4) {
        v2f a, b;
        a.x = pf[k0]; a.y = pf[k0 + 1];            // A: M=m, K=k0+2*khalf (+1)
        b.x = pc[k0]; b.y = pc[k0 + 1];            // B: N=m, same K slice
        sqf += (double)a.x * a.x + (double)a.y * a.y;
        sqc += (double)b.x * b.x + (double)b.y * b.y;
        // 8-arg form: (neg_a, A, neg_b, B, c_mod, C, reuse_a, reuse_b)
        acc = __builtin_amdgcn_wmma_f32_16x16x4_f32(
            false, a, false, b, (short)0, acc, false, false);
    }
    // lane m and lane m+16 each hold half of row m's f64 sums -> combine
    sqf += __shfl_xor(sqf, 16, 32);
    sqc += __shfl_xor(sqc, 16, 32);

    // D layout (32-bit C/D 16x16): VGPR r, lanes 0-15 hold M=r, lanes 16-31
    // hold M=r+8. Diagonal (m,m): m<8 -> lane m, acc[m]; m>=8 -> lane m+16,
    // acc[m-8]. So lanes 0-7 and 24-31 each own one diagonal element acc[lane&7].
    const int s = lane & 7;
    float dot;
    switch (s) {
        case 0: dot = acc[0]; break;
        case 1: dot = acc[1]; break;
        case 2: dot = acc[2]; break;
        case 3: dot = acc[3]; break;
        case 4: dot = acc[4]; break;
        case 5: dot = acc[5]; break;
        case 6: dot = acc[6]; break;
        default: dot = acc[7]; break;
    }
    if (lane < 8 || lane >= 24) {
        const int row = (lane < 8) ? lane : (lane - 16);   // sq sums match: m==row
        double dist = sqf + sqc - 2.0 * (double)dot;
        dist = dist < 1e-12 ? 1e-12 : (dist > 1e12 ? 1e12 : dist);
        partial[n0 + row] = dist;
    }
}

// ---------------------------------------------------------------------------
// Kernel 3: deterministic final reduction of N=10240 f64 partials, add the
// clip-floor constant from the (C-1) masked-out entries per row, mean over N.
// ---------------------------------------------------------------------------
__global__ __launch_bounds__(1024) void cl_reduce_kernel(
    const double* __restrict__ partial, float* __restrict__ out) {
    __shared__ double sm[1024];
    const int t = threadIdx.x;
    double s = 0.0;
    for (int k = t; k < N_; k += 1024) s += partial[k];   // fixed order
    sm[t] = s;
    __syncthreads();
    for (int st = 512; st > 0; st >>= 1) {
        if (t < st) sm[t] += sm[t + st];
        __syncthreads();
    }
    if (t == 0) {
        double total = sm[0] + (double)N_ * (double)(C_ - 1) * 1e-12;
        out[0] = (float)(total / (double)N_);
    }
}

extern "C" void kernel_launch(void* const* d_in, const int* in_sizes, int n_in,
                              void* d_out, int out_size, void* d_ws, size_t ws_size,
                              hipStream_t stream) {
    (void)in_sizes; (void)n_in; (void)out_size; (void)ws_size;
    const float* logits  = (const float*)d_in[0];   // [B,T,C] f32
    const float* feats   = (const float*)d_in[1];   // [B,T,D] f32
    const float* centers = (const float*)d_in[2];   // [C,D]   f32

    int*    labels  = (int*)d_ws;
    double* partial = (double*)((char*)d_ws + ((N_ * sizeof(int) + 255) & ~(size_t)255));

    cl_argmax_kernel   <<<N_ / 8,        256,  0, stream>>>(logits, labels);
    cl_dist_wmma_kernel<<<(N_ / 16) / 8, 256,  0, stream>>>(feats, centers, labels, partial);
    cl_reduce_kernel   <<<1,             1024, 0, stream>>>(partial, (float*)d_out);
}